// VMD_8933531975739
// MI455X (gfx1250) — compile-verified
//
#include <hip/hip_runtime.h>
#include <hip/hip_bf16.h>
#include <math.h>

// ---------------------------------------------------------------------------
// VMD (8,2048,32), K=3, 100 iters — frequency-domain formulation.
// State per (b,d) channel kept in LDS; one 2048-pt FFT per iteration,
// implemented as four-step 64x32 decomposition with V_WMMA_F32_16X16X4_F32.
// ---------------------------------------------------------------------------

#define PI_F 3.14159265358979323846f

typedef __attribute__((ext_vector_type(2))) float v2f;
typedef __attribute__((ext_vector_type(8))) float v8f;

// One K=4 WMMA FMA step: C += A(16x4) * B(4x16), fp32.
__device__ __forceinline__ v8f wmma4(v2f a, v2f b, v8f c) {
  return __builtin_amdgcn_wmma_f32_16x16x4_f32(
      /*neg_a=*/false, a, /*neg_b=*/false, b,
      /*c_mod=*/(short)0, c, /*reuse_a=*/false, /*reuse_b=*/false);
}

// In-LDS 2048-point complex forward FFT (W = e^{-2*pi*i/N}), natural order
// in and out.  Four-step: N1=64 rows, N2=32 cols.
//   A[n1][n2] = io[n1*32+n2]
//   B = D64 * A              (stage 1, WMMA)
//   C[k1][n2] = B * W_2048^{k1*n2}   (stage 2, VALU twiddle)
//   X = C * D32; io[k1 + 64*k2] = X[k1][k2]  (stage 3, WMMA)
// Requires 256 threads (8 waves); each wave owns one 16x16 output tile.
__device__ void fft2048(float* io_re, float* io_im,
                        float* s_re,  float* s_im,
                        const float* d64r, const float* d64i,
                        const float* d32r, const float* d32i) {
  const int t    = threadIdx.x;
  const int lane = t & 31;
  const int w    = t >> 5;        // wave 0..7
  const int rt   = w >> 1;        // row tile 0..3 (k1 / 16)
  const int ct   = w & 1;         // col tile 0..1
  const int half = lane >> 4;     // lane half (0/1)
  const int l15  = lane & 15;

  // ---- Stage 1: B(64x32) = D64(64x64) * A(64x32) --------------------------
  {
    v8f cr = {};
    v8f ci = {};
    const int arow = rt * 16 + l15;
    const int bcol = ct * 16 + l15;
#pragma unroll
    for (int kk = 0; kk < 16; ++kk) {
      const int ka = 4 * kk + 2 * half;
      v2f dR, dI, bR, bI, nI;
      dR.x = d64r[arow * 64 + ka];     dR.y = d64r[arow * 64 + ka + 1];
      dI.x = d64i[arow * 64 + ka];     dI.y = d64i[arow * 64 + ka + 1];
      bR.x = io_re[(ka    ) * 32 + bcol];
      bR.y = io_re[(ka + 1) * 32 + bcol];
      bI.x = io_im[(ka    ) * 32 + bcol];
      bI.y = io_im[(ka + 1) * 32 + bcol];
      nI.x = -bI.x; nI.y = -bI.y;
      cr = wmma4(dR, bR, cr);   // + Dr*Ar
      cr = wmma4(dI, nI, cr);   // - Di*Ai
      ci = wmma4(dR, bI, ci);   // + Dr*Ai
      ci = wmma4(dI, bR, ci);   // + Di*Ar
    }
    __syncthreads();            // all stage-1 reads of io complete
#pragma unroll
    for (int v = 0; v < 8; ++v) {
      const int row = rt * 16 + v + 8 * half;
      const int col = ct * 16 + l15;
      s_re[row * 32 + col] = cr[v];
      s_im[row * 32 + col] = ci[v];
    }
  }
  __syncthreads();

  // ---- Stage 2: twiddle C[k1][n2] *= exp(-2*pi*i*k1*n2/2048) --------------
#pragma unroll
  for (int e = 0; e < 8; ++e) {
    const int l  = t + 256 * e;
    const int k1 = l >> 5;
    const int n2 = l & 31;
    float sn, cs;
    __sincosf(-2.0f * PI_F * (float)(k1 * n2) * (1.0f / 2048.0f), &sn, &cs);
    const float re = s_re[l], im = s_im[l];
    s_re[l] = re * cs - im * sn;
    s_im[l] = re * sn + im * cs;
  }
  __syncthreads();

  // ---- Stage 3: X(64x32) = C(64x32) * D32(32x32); io[k1 + 64*k2] = X ------
  {
    v8f cr = {};
    v8f ci = {};
    const int arow = rt * 16 + l15;
    const int bcol = ct * 16 + l15;
#pragma unroll
    for (int kk = 0; kk < 8; ++kk) {
      const int ka = 4 * kk + 2 * half;
      v2f aR, aI, dR, dI, nI;
      aR.x = s_re[arow * 32 + ka];     aR.y = s_re[arow * 32 + ka + 1];
      aI.x = s_im[arow * 32 + ka];     aI.y = s_im[arow * 32 + ka + 1];
      dR.x = d32r[(ka    ) * 32 + bcol];
      dR.y = d32r[(ka + 1) * 32 + bcol];
      dI.x = d32i[(ka    ) * 32 + bcol];
      dI.y = d32i[(ka + 1) * 32 + bcol];
      nI.x = -dI.x; nI.y = -dI.y;
      cr = wmma4(aR, dR, cr);   // + Cr*Dr
      cr = wmma4(aI, nI, cr);   // - Ci*Di
      ci = wmma4(aR, dI, ci);   // + Cr*Di
      ci = wmma4(aI, dR, ci);   // + Ci*Dr
    }
#pragma unroll
    for (int v = 0; v < 8; ++v) {
      const int k1 = rt * 16 + v + 8 * half;
      const int k2 = ct * 16 + l15;
      io_re[k1 + 64 * k2] = cr[v];
      io_im[k1 + 64 * k2] = ci[v];
    }
  }
  __syncthreads();
}

// ---------------------------------------------------------------------------
// Main VMD kernel: one block per (b,d) channel. 256 threads, dynamic LDS.
// ---------------------------------------------------------------------------
__global__ void __launch_bounds__(256)
vmd_main(const float* __restrict__ x, float* __restrict__ ws_u,
         float* __restrict__ ws_omega) {
  extern __shared__ float sm[];
  float* xhat_re = sm;                 // 2048
  float* xhat_im = xhat_re + 2048;     // 2048
  float* uh_re   = xhat_im + 2048;     // 3*2048 (fftshifted spectra of modes)
  float* uh_im   = uh_re + 6144;       // 3*2048
  float* lam     = uh_im + 6144;       // 2048
  float* xcol    = lam + 2048;         // 2048
  float* fio_re  = xcol + 2048;        // 2048 FFT in/out
  float* fio_im  = fio_re + 2048;      // 2048
  float* fs_re   = fio_im + 2048;      // 2048 FFT scratch / temp spectrum
  float* fs_im   = fs_re + 2048;       // 2048
  float* d64r    = fs_im + 2048;       // 4096 DFT64 tables
  float* d64i    = d64r + 4096;        // 4096
  float* d32r    = d64i + 4096;        // 1024 DFT32 tables
  float* d32i    = d32r + 1024;        // 1024
  float* redn    = d32i + 1024;        // 256 reduction scratch
  float* redd    = redn + 256;         // 256
  float* omg     = redd + 256;         // 4

  const int t = threadIdx.x;
  const int col = blockIdx.x;          // 0..255
  const int b = col >> 5;
  const int d = col & 31;

  // DFT tables: W_64^{k*n}, W_32^{k*n}
  for (int i = t; i < 4096; i += 256) {
    const int k = i >> 6, n = i & 63;
    float sn, cs;
    __sincosf(-2.0f * PI_F * (float)(k * n) * (1.0f / 64.0f), &sn, &cs);
    d64r[i] = cs; d64i[i] = sn;
  }
  for (int i = t; i < 1024; i += 256) {
    const int k = i >> 5, n = i & 31;
    float sn, cs;
    __sincosf(-2.0f * PI_F * (float)(k * n) * (1.0f / 32.0f), &sn, &cs);
    d32r[i] = cs; d32i[i] = sn;
  }
  // Load channel, zero state.
  for (int l = t; l < 2048; l += 256) {
    const float xv = x[((size_t)b * 2048 + l) * 32 + d];
    xcol[l] = xv;
    fio_re[l] = xv; fio_im[l] = 0.0f;
    lam[l] = 0.0f;
    uh_re[l] = uh_re[2048 + l] = uh_re[4096 + l] = 0.0f;
    uh_im[l] = uh_im[2048 + l] = uh_im[4096 + l] = 0.0f;
  }
  if (t < 4) omg[t] = 0.0f;
  __syncthreads();

  // x_hat = fftshift(fft(x))
  fft2048(fio_re, fio_im, fs_re, fs_im, d64r, d64i, d32r, d32i);
  for (int l = t; l < 2048; l += 256) {
    const int m = (l + 1024) & 2047;
    xhat_re[l] = fio_re[m];
    xhat_im[l] = fio_im[m];
  }
  __syncthreads();

  for (int it = 0; it < 100; ++it) {
    // ---- sequential mode updates (frequency domain) ----
    for (int k = 0; k < 3; ++k) {
      const int ka = (k + 1 == 3) ? 0 : k + 1;
      const int kb = (k + 2 >= 3) ? k - 1 : k + 2;
      const float om = omg[k];
      for (int l = t; l < 2048; l += 256) {
        const float sor = uh_re[ka * 2048 + l] + uh_re[kb * 2048 + l];
        const float soi = uh_im[ka * 2048 + l] + uh_im[kb * 2048 + l];
        const float nr = xhat_re[l] - sor + 0.5f * lam[l];
        const float ni = xhat_im[l] - soi;
        const float f  = ((float)l - 1024.0f) * (1.0f / 2048.0f);
        const float df = f - om;
        const float inv = 1.0f / (1.0f + 2.0f * 2000.0f * df * df);
        fs_re[l] = nr * inv;
        fs_im[l] = ni * inv;
      }
      __syncthreads();
      // Hermitian projection == fft(real(ifft(.))) in shifted index space
      for (int l = t; l < 2048; l += 256) {
        const int lp = (2048 - l) & 2047;
        uh_re[k * 2048 + l] = 0.5f * (fs_re[l] + fs_re[lp]);
        uh_im[k * 2048 + l] = 0.5f * (fs_im[l] - fs_im[lp]);
      }
      __syncthreads();
    }

    // ---- omega update: centroid of |uhat|^2 over f >= 0 (l >= 1024) ----
    for (int k = 0; k < 3; ++k) {
      float pn = 0.0f, pd = 0.0f;
      for (int l = 1024 + t; l < 2048; l += 256) {
        const float re = uh_re[k * 2048 + l], im = uh_im[k * 2048 + l];
        const float p = re * re + im * im;
        const float f = ((float)l - 1024.0f) * (1.0f / 2048.0f);
        pn += f * p; pd += p;
      }
      redn[t] = pn; redd[t] = pd;
      __syncthreads();
      for (int s = 128; s > 0; s >>= 1) {
        if (t < s) { redn[t] += redn[t + s]; redd[t] += redd[t + s]; }
        __syncthreads();
      }
      if (t == 0) omg[k] = redn[0] / (redd[0] + 1e-7f);
      __syncthreads();
    }

    // ---- lambda update: needs time-domain sum of modes (ONE ifft) ----
    // ifft(ifftshift(S)) = conj(FFT(conj(ifftshift(S)))) / N ; real part used.
    for (int l = t; l < 2048; l += 256) {
      const int m = (l + 1024) & 2047;
      fio_re[l] =  (uh_re[m] + uh_re[2048 + m] + uh_re[4096 + m]);
      fio_im[l] = -(uh_im[m] + uh_im[2048 + m] + uh_im[4096 + m]);
    }
    __syncthreads();
    fft2048(fio_re, fio_im, fs_re, fs_im, d64r, d64i, d32r, d32i);
    for (int l = t; l < 2048; l += 256) {
      const float usum = fio_re[l] * (1.0f / 2048.0f);
      lam[l] += 0.001f * (xcol[l] - usum);
    }
    __syncthreads();
  }

  // ---- epilogue: modes back to time domain, store unsorted + omega ----
  for (int k = 0; k < 3; ++k) {
    for (int l = t; l < 2048; l += 256) {
      const int m = (l + 1024) & 2047;
      fio_re[l] =  uh_re[k * 2048 + m];
      fio_im[l] = -uh_im[k * 2048 + m];
    }
    __syncthreads();
    fft2048(fio_re, fio_im, fs_re, fs_im, d64r, d64i, d32r, d32i);
    for (int l = t; l < 2048; l += 256) {
      ws_u[(((size_t)b * 3 + k) * 2048 + l) * 32 + d] =
          fio_re[l] * (1.0f / 2048.0f);
    }
    __syncthreads();
  }
  if (t == 0) {
    ws_omega[(b * 3 + 0) * 32 + d] = omg[0];
    ws_omega[(b * 3 + 1) * 32 + d] = omg[1];
    ws_omega[(b * 3 + 2) * 32 + d] = omg[2];
  }
}

// ---------------------------------------------------------------------------
// Per-batch: mean omega over d, stable argsort of 3.
// ---------------------------------------------------------------------------
__global__ void vmd_order(const float* __restrict__ ws_omega,
                          int* __restrict__ ws_order) {
  const int b = blockIdx.x;
  const int d = threadIdx.x;  // 32 lanes (one wave32)
  float m[3];
  for (int k = 0; k < 3; ++k) {
    float v = ws_omega[(b * 3 + k) * 32 + d];
    for (int off = 16; off > 0; off >>= 1) v += __shfl_down(v, off, 32);
    m[k] = v;  // lane 0 holds the sum (mean = sum/32; monotonic, skip divide)
  }
  if (d == 0) {
    int idx[3] = {0, 1, 2};
    float vv[3] = {m[0], m[1], m[2]};
    for (int i = 0; i < 2; ++i) {           // stable selection (strict <)
      int best = i;
      for (int j = i + 1; j < 3; ++j)
        if (vv[j] < vv[best]) best = j;
      const float tv = vv[best]; const int ti = idx[best];
      for (int j = best; j > i; --j) { vv[j] = vv[j - 1]; idx[j] = idx[j - 1]; }
      vv[i] = tv; idx[i] = ti;
    }
    ws_order[b * 4 + 0] = idx[0];
    ws_order[b * 4 + 1] = idx[1];
    ws_order[b * 4 + 2] = idx[2];
  }
}

// ---------------------------------------------------------------------------
// Gather sorted modes into d_out: out[k][b][l][d] = ws_u[b][order[b][k]][l][d]
// ---------------------------------------------------------------------------
__global__ void vmd_gather(const float* __restrict__ ws_u,
                           const int* __restrict__ ws_order,
                           float* __restrict__ out) {
  const size_t i = (size_t)blockIdx.x * blockDim.x + threadIdx.x;
  const int dd = (int)(i & 31);
  size_t r = i >> 5;
  const int l = (int)(r & 2047); r >>= 11;
  const int b = (int)(r & 7);
  const int k = (int)(r >> 3);
  const int ks = ws_order[b * 4 + k];
  out[i] = ws_u[(((size_t)b * 3 + ks) * 2048 + l) * 32 + dd];
}

// ---------------------------------------------------------------------------
extern "C" void kernel_launch(void* const* d_in, const int* in_sizes, int n_in,
                              void* d_out, int out_size, void* d_ws,
                              size_t ws_size, hipStream_t stream) {
  const float* x = (const float*)d_in[0];
  float* out = (float*)d_out;

  // Workspace layout (floats): u_unsorted [8][3][2048][32], omega [8][3][32],
  // order [8][4] ints.  ~6.3 MB total.
  float* ws_u = (float*)d_ws;
  float* ws_omega = ws_u + (size_t)8 * 3 * 2048 * 32;
  int* ws_order = (int*)(ws_omega + 8 * 3 * 32);

  constexpr size_t SMEM_FLOATS =
      2048 * 2 +      // xhat
      6144 * 2 +      // uhat
      2048 +          // lam
      2048 +          // xcol
      2048 * 2 +      // fft io
      2048 * 2 +      // fft scratch
      4096 * 2 +      // DFT64
      1024 * 2 +      // DFT32
      256 * 2 +       // reduction
      4;              // omega
  const size_t smem_bytes = SMEM_FLOATS * sizeof(float);  // ~154 KB < 320 KB

  vmd_main<<<256, 256, smem_bytes, stream>>>(x, ws_u, ws_omega);
  vmd_order<<<8, 32, 0, stream>>>(ws_omega, ws_order);
  vmd_gather<<<(3 * 8 * 2048 * 32) / 256, 256, 0, stream>>>(ws_u, ws_order,
                                                            out);
}